// KmeansAttention_74663711474289
// MI455X (gfx1250) — compile-verified
//
#include <hip/hip_runtime.h>
#include <math.h>

// ---------------- problem constants (from reference) ----------------
#define B 2
#define H 8
#define T 8192
#define D 64
#define C 64
#define WSZ 128
#define NC (T / WSZ)          // 64 windows
#define SCALE 0.125f          // d^-0.5
#define TOKEN_SELF -50000.0f

#define QS_STRIDE 66          // f32 LDS row stride: (2*row + k) % 64 -> conflict-free frags
#define PB_STRIDE 130
#define DB_STRIDE 65

#if defined(__has_builtin)
#if __has_builtin(__builtin_amdgcn_global_load_async_to_lds_b128) && \
    __has_builtin(__builtin_amdgcn_s_wait_asynccnt)
#define HAVE_ASYNC_LDS 1
#endif
#endif

typedef float v2f __attribute__((ext_vector_type(2)));
typedef float v8f __attribute__((ext_vector_type(8)));
typedef int   v4i __attribute__((ext_vector_type(4)));
typedef v4i __attribute__((address_space(1))) gas_v4i;   // global
typedef v4i __attribute__((address_space(3))) las_v4i;   // LDS

// D = A(16x4,f32) * B(4x16,f32) + C(16x16,f32), wave32.
// A: lane m = lane%16; halfwave 0 holds K=k0,k0+1, halfwave 1 holds K=k0+2,k0+3.
// B mirrored: lane n = lane%16, same K split. C: c[r] is (M = r + 8*(lane/16), N = lane%16).
static __device__ __forceinline__ v8f wmma_f32_k4(float ax, float ay,
                                                  float bx, float by, v8f c) {
  v2f a; a.x = ax; a.y = ay;
  v2f b; b.x = bx; b.y = by;
  return __builtin_amdgcn_wmma_f32_16x16x4_f32(false, a, false, b, (short)0, c,
                                               false, false);
}

static __device__ __forceinline__ void atomic_add_f32(float* p, float v) {
  __hip_atomic_fetch_add(p, v, __ATOMIC_RELAXED, __HIP_MEMORY_SCOPE_AGENT);
}

// monotone map: larger float -> larger uint (no NaNs in dists)
static __device__ __forceinline__ unsigned mapU(unsigned x) {
  return (x & 0x80000000u) ? ~x : (x | 0x80000000u);
}

// ==================== K1: routing similarities ====================
// dists[b,h,c,t] = (qk[b,h,t,:] . means[h,c,:]) / max(||qk||, 1e-12)
__global__ __launch_bounds__(256) void k_dists(const float* __restrict__ qk,
                                               const float* __restrict__ means,
                                               float* __restrict__ dists) {
  __shared__ float qs[WSZ][QS_STRIDE];
  __shared__ float ms[C][QS_STRIDE];
  __shared__ float rn[WSZ];
  __shared__ float db[WSZ][DB_STRIDE];   // staging for coalesced [c][t] store

  const int tid = threadIdx.x;
  const int bh  = blockIdx.x / (T / WSZ);
  const int tb  = blockIdx.x % (T / WSZ);
  const int h   = bh % H;

  {  // 128 qk rows, 2 threads/row, float4 global reads
    const float4* src = (const float4*)(qk + ((size_t)bh * T + (size_t)tb * WSZ) * D);
    int r = tid >> 1, half = tid & 1;
    for (int u = 0; u < 8; ++u) {
      float4 x = src[(size_t)r * (D / 4) + half * 8 + u];
      int c0 = (half * 8 + u) * 4;
      qs[r][c0] = x.x; qs[r][c0 + 1] = x.y; qs[r][c0 + 2] = x.z; qs[r][c0 + 3] = x.w;
    }
  }
  {  // means for this h (already l2-normalized by setup)
    const float4* src = (const float4*)(means + (size_t)h * C * D);
    for (int i = tid; i < C * (D / 4); i += 256) {
      float4 x = src[i];
      int r = i >> 4, c0 = (i & 15) * 4;
      ms[r][c0] = x.x; ms[r][c0 + 1] = x.y; ms[r][c0 + 2] = x.z; ms[r][c0 + 3] = x.w;
    }
  }
  __syncthreads();
  if (tid < WSZ) {
    float s = 0.f;
    for (int d = 0; d < D; ++d) { float q = qs[tid][d]; s += q * q; }
    rn[tid] = fmaxf(sqrtf(s), 1e-12f);
  }
  __syncthreads();

  const int lane = tid & 31, ln = lane & 15, hi = lane >> 4;
  const int m0 = (tid >> 5) * 16;
  for (int ct = 0; ct < 4; ++ct) {           // 64 clusters = 4 N-tiles
    v8f acc = {0, 0, 0, 0, 0, 0, 0, 0};
    for (int kk = 0; kk < 16; ++kk) {        // K = 64
      int k0 = kk * 4 + 2 * hi;
      acc = wmma_f32_k4(qs[m0 + ln][k0], qs[m0 + ln][k0 + 1],
                        ms[ct * 16 + ln][k0], ms[ct * 16 + ln][k0 + 1], acc);
    }
    for (int r = 0; r < 8; ++r) {
      int m = m0 + r + 8 * hi;
      db[m][ct * 16 + ln] = acc[r] / rn[m];
    }
  }
  __syncthreads();
  {
    float* dst = dists + (size_t)bh * C * T + (size_t)tb * WSZ;
    for (int i = tid; i < C * WSZ; i += 256) {
      int c = i >> 7, tok = i & 127;
      dst[(size_t)c * T + tok] = db[tok][c];
    }
  }
}

// ==================== K2: top-128 per cluster via radix select ====================
// MSB-first 8-bit radix select over order-preserving uint keys -> exact 32-bit
// pivot + tie budget (lowest indices first, matching jax.lax.top_k), then a
// stable index-ordered compaction so outputs are already ascending (jnp.sort).
__global__ __launch_bounds__(256) void k_topk(const float* __restrict__ dists,
                                              int* __restrict__ idxout) {
  __shared__ unsigned su[T];        // 32KB: raw bits, then mapped keys
  __shared__ unsigned hist[256];
  __shared__ unsigned s_cs[256];    // per-thread strict-greater counts -> excl scan
  __shared__ unsigned s_ct[256];    // per-thread tie counts            -> excl scan
  __shared__ unsigned sh_pfx, sh_need;

  const int tid = threadIdx.x;
  const float* src = dists + (size_t)blockIdx.x * T;   // blockIdx = bh*C + c

#if HAVE_ASYNC_LDS
  // contiguous 32KB row: async copy global -> LDS, 16B granules, coalesced
  for (int k = 0; k < 8; ++k) {
    int i4 = tid + k * 256;                             // float4 index
    __builtin_amdgcn_global_load_async_to_lds_b128(
        (gas_v4i*)(src + (size_t)i4 * 4),
        (las_v4i*)&su[i4 * 4], 0, 0);
  }
  __builtin_amdgcn_s_wait_asynccnt(0);
#else
  {
    const float4* s4 = (const float4*)src;
    for (int k = 0; k < 8; ++k) {
      int i4 = tid + k * 256;
      float4 x = s4[i4];
      int b0 = i4 * 4;
      su[b0]     = __float_as_uint(x.x);
      su[b0 + 1] = __float_as_uint(x.y);
      su[b0 + 2] = __float_as_uint(x.z);
      su[b0 + 3] = __float_as_uint(x.w);
    }
  }
#endif
  if (tid == 0) { sh_pfx = 0u; sh_need = WSZ; }
  __syncthreads();

  for (int i = tid; i < T; i += 256) su[i] = mapU(su[i]);   // in-place, disjoint
  __syncthreads();

  // 4 radix levels, MSB first
  for (int level = 0; level < 4; ++level) {
    hist[tid] = 0u;
    __syncthreads();
    const unsigned pfx = sh_pfx;
    const int shift = 24 - 8 * level;
    for (int i = tid; i < T; i += 256) {
      unsigned u = su[i];
      if (level == 0 || (u >> (shift + 8)) == pfx)
        atomicAdd(&hist[(u >> shift) & 255u], 1u);
    }
    __syncthreads();
    if (tid == 0) {
      unsigned need = sh_need, cum = 0;
      for (int b = 255; b >= 0; --b) {
        unsigned c = hist[b];
        if (cum + c >= need) { sh_pfx = (pfx << 8) | (unsigned)b; sh_need = need - cum; break; }
        cum += c;
      }
    }
    __syncthreads();
  }

  const unsigned pivot = sh_pfx;        // exact 32-bit key of the 128th element
  const unsigned tiesNeeded = sh_need;  // #pivot-equal elems to keep (lowest idx)

  // per-thread counts over its contiguous 32 indices
  const int base = tid * 32;
  unsigned cs = 0, ctt = 0;
  for (int j = 0; j < 32; ++j) {
    unsigned u = su[base + j];
    cs  += (u > pivot);
    ctt += (u == pivot);
  }
  s_cs[tid] = cs; s_ct[tid] = ctt;
  __syncthreads();
  if (tid == 0) {   // serial exclusive scans (256 elems, trivial cost)
    unsigned a = 0, b = 0;
    for (int i = 0; i < 256; ++i) {
      unsigned ta = s_cs[i], tb = s_ct[i];
      s_cs[i] = a; s_ct[i] = b;
      a += ta; b += tb;
    }
  }
  __syncthreads();

  // stable compaction in index order: output is ascending automatically
  unsigned sb = s_cs[tid], tb = s_ct[tid];
  int* dst = idxout + (size_t)blockIdx.x * WSZ;
  for (int j = 0; j < 32; ++j) {
    int idx = base + j;
    unsigned u = su[idx];
    if (u > pivot) {
      unsigned pos = sb + (tb < tiesNeeded ? tb : tiesNeeded);
      dst[pos] = idx;
      ++sb;
    } else if (u == pivot) {
      if (tb < tiesNeeded) dst[sb + tb] = idx;
      ++tb;
    }
  }
}

// ==================== K3: windowed attention + scatter ====================
__global__ __launch_bounds__(256) void k_attn(const float* __restrict__ qk,
                                              const float* __restrict__ v,
                                              const float* __restrict__ relw,
                                              const int* __restrict__ idxbuf,
                                              float* __restrict__ out,
                                              float* __restrict__ denom) {
  __shared__ int   sidx[WSZ];
  __shared__ float qs[WSZ][QS_STRIDE];   // gathered qk, later gathered v
  __shared__ float ks[WSZ][QS_STRIDE];   // relW stage, later l2norm(k)
  __shared__ float pb[WSZ][PB_STRIDE];   // p = scale*q@relW^T, later attn

  const int tid = threadIdx.x;
  const int bh  = blockIdx.x / NC;
  const int n   = blockIdx.x % NC;
  const int h   = bh % H;

  if (tid < WSZ) sidx[tid] = idxbuf[((size_t)bh * NC + n) * WSZ + tid];
  __syncthreads();

  const int r2 = tid >> 1, half = tid & 1;
  {  // gather q rows + stage relW rows for this head
    const float4* qb = (const float4*)qk;
    size_t rowbase = ((size_t)bh * T + (size_t)sidx[r2]) * (D / 4);
    for (int u = 0; u < 8; ++u) {
      float4 x = qb[rowbase + half * 8 + u];
      int c0 = (half * 8 + u) * 4;
      qs[r2][c0] = x.x; qs[r2][c0 + 1] = x.y; qs[r2][c0 + 2] = x.z; qs[r2][c0 + 3] = x.w;
    }
    const float4* rb = (const float4*)relw;
    size_t rrow = ((size_t)r2 * H + h) * (D / 4);
    for (int u = 0; u < 8; ++u) {
      float4 x = rb[rrow + half * 8 + u];
      int c0 = (half * 8 + u) * 4;
      ks[r2][c0] = x.x; ks[r2][c0 + 1] = x.y; ks[r2][c0 + 2] = x.z; ks[r2][c0 + 3] = x.w;
    }
  }
  __syncthreads();

  const int lane = tid & 31, ln = lane & 15, hi = lane >> 4;
  const int m0 = (tid >> 5) * 16;

  // p[m][j] = scale * sum_d q[m,d]*relW[j,d]  (each wave writes its own 16 rows)
  for (int ct = 0; ct < 8; ++ct) {
    v8f acc = {0, 0, 0, 0, 0, 0, 0, 0};
    for (int kk = 0; kk < 16; ++kk) {
      int k0 = kk * 4 + 2 * hi;
      acc = wmma_f32_k4(qs[m0 + ln][k0], qs[m0 + ln][k0 + 1],
                        ks[ct * 16 + ln][k0], ks[ct * 16 + ln][k0 + 1], acc);
    }
    for (int r = 0; r < 8; ++r)
      pb[m0 + r + 8 * hi][ct * 16 + ln] = acc[r] * SCALE;
  }
  __syncthreads();   // everyone done reading relW from ks

  if (tid < WSZ) {   // ks = l2norm(gathered q)
    float s = 0.f;
    for (int d = 0; d < D; ++d) { float q = qs[tid][d]; s += q * q; }
    float inv = 1.0f / fmaxf(sqrtf(s), 1e-12f);
    for (int d = 0; d < D; ++d) ks[tid][d] = qs[tid][d] * inv;
  }
  __syncthreads();

  // dots + shifted rel + self mask, kept in registers
  v8f accs[8];
  for (int ct = 0; ct < 8; ++ct) {
    v8f acc = {0, 0, 0, 0, 0, 0, 0, 0};
    for (int kk = 0; kk < 16; ++kk) {
      int k0 = kk * 4 + 2 * hi;
      acc = wmma_f32_k4(qs[m0 + ln][k0], qs[m0 + ln][k0 + 1],
                        ks[ct * 16 + ln][k0], ks[ct * 16 + ln][k0 + 1], acc);
    }
    for (int r = 0; r < 8; ++r) {
      int m = m0 + r + 8 * hi;
      int nG = ct * 16 + ln;
      float val = acc[r] * SCALE;
      if (nG == m)      val = TOKEN_SELF;            // eye mask
      else if (nG < m)  val += pb[m][127 + nG - m];  // shift(): rel for n<m, else 0
      acc[r] = val;
    }
    accs[ct] = acc;
  }

  // row softmax: reduce over 8 N-tiles + 16-lane shfl groups
  for (int r = 0; r < 8; ++r) {
    float mx = -3.0e38f;
    for (int ct = 0; ct < 8; ++ct) mx = fmaxf(mx, accs[ct][r]);
    for (int off = 1; off < 16; off <<= 1) mx = fmaxf(mx, __shfl_xor(mx, off, 32));
    float sum = 0.f;
    for (int ct = 0; ct < 8; ++ct) {
      float e = expf(accs[ct][r] - mx);
      accs[ct][r] = e; sum += e;
    }
    for (int off = 1; off < 16; off <<= 1) sum += __shfl_xor(sum, off, 32);
    float inv = 1.0f / sum;
    for (int ct = 0; ct < 8; ++ct) accs[ct][r] *= inv;
  }

  // attn -> pb (own rows only; this wave's rel reads of pb are already done)
  for (int ct = 0; ct < 8; ++ct)
    for (int r = 0; r < 8; ++r)
      pb[m0 + r + 8 * hi][ct * 16 + ln] = accs[ct][r];
  __syncthreads();   // qs about to be repurposed for v

  {  // gather v rows into qs
    const float4* vb = (const float4*)v;
    size_t rowbase = ((size_t)bh * T + (size_t)sidx[r2]) * (D / 4);
    for (int u = 0; u < 8; ++u) {
      float4 x = vb[rowbase + half * 8 + u];
      int c0 = (half * 8 + u) * 4;
      qs[r2][c0] = x.x; qs[r2][c0 + 1] = x.y; qs[r2][c0 + 2] = x.z; qs[r2][c0 + 3] = x.w;
    }
  }
  __syncthreads();

  // bo = attn @ v (K = 128), scatter-add to original token order
  for (int ct2 = 0; ct2 < 4; ++ct2) {
    v8f acc = {0, 0, 0, 0, 0, 0, 0, 0};
    for (int kk = 0; kk < 32; ++kk) {
      int k0 = kk * 4 + 2 * hi;
      acc = wmma_f32_k4(pb[m0 + ln][k0], pb[m0 + ln][k0 + 1],
                        qs[k0][ct2 * 16 + ln], qs[k0 + 1][ct2 * 16 + ln], acc);
    }
    for (int r = 0; r < 8; ++r) {
      int m = m0 + r + 8 * hi;
      int tok = sidx[m];
      atomic_add_f32(&out[((size_t)bh * T + tok) * D + ct2 * 16 + ln], acc[r]);
    }
  }
  if (tid < WSZ)
    atomic_add_f32(&denom[(size_t)bh * T + sidx[tid]], 1.0f);
}

// ==================== K4: scatter-mean finalize ====================
__global__ __launch_bounds__(256) void k_final(float* __restrict__ out,
                                               const float* __restrict__ denom) {
  size_t i = (size_t)blockIdx.x * blockDim.x + threadIdx.x;   // float4 index
  const size_t n4 = (size_t)B * H * T * D / 4;
  if (i >= n4) return;
  float4* o4 = (float4*)out;
  float4 x = o4[i];
  float inv = 1.0f / (denom[i / (D / 4)] + 1e-5f);
  x.x *= inv; x.y *= inv; x.z *= inv; x.w *= inv;
  o4[i] = x;
}

// ==================== host launch ====================
extern "C" void kernel_launch(void* const* d_in, const int* in_sizes, int n_in,
                              void* d_out, int out_size, void* d_ws, size_t ws_size,
                              hipStream_t stream) {
  const float* qk    = (const float*)d_in[0];
  const float* v     = (const float*)d_in[1];
  const float* means = (const float*)d_in[2];
  const float* relw  = (const float*)d_in[3];
  float* out = (float*)d_out;

  // ws layout: dists (33.5 MB) | idx (512 KB) | denom (512 KB)  ~= 34.6 MB
  const size_t distBytes = (size_t)B * H * C * T * sizeof(float);
  const size_t idxBytes  = (size_t)B * H * C * WSZ * sizeof(int);
  float* dists = (float*)d_ws;
  int*   idxb  = (int*)((char*)d_ws + distBytes);
  float* denom = (float*)((char*)d_ws + distBytes + idxBytes);

  (void)hipMemsetAsync(d_out, 0, (size_t)out_size * sizeof(float), stream);
  (void)hipMemsetAsync(denom, 0, (size_t)B * H * T * sizeof(float), stream);

  k_dists<<<B * H * (T / WSZ), 256, 0, stream>>>(qk, means, dists);
  k_topk <<<B * H * C,         256, 0, stream>>>(dists, idxb);
  k_attn <<<B * H * NC,        256, 0, stream>>>(qk, v, relw, idxb, out, denom);
  const int n4 = B * H * T * D / 4;
  k_final<<<(n4 + 255) / 256,  256, 0, stream>>>(out, denom);
}